// ProjectedGaussianRasterizer_18090402251359
// MI455X (gfx1250) — compile-verified
//
#include <hip/hip_runtime.h>
#include <hip/hip_bf16.h>

typedef __attribute__((ext_vector_type(2))) float v2f;
typedef __attribute__((ext_vector_type(4))) float v4f;
typedef __attribute__((ext_vector_type(8))) float v8f;

#define HH 128
#define WW 128
#define GG 2048

__device__ __forceinline__ float shflx16(float v) {
    return __shfl_xor(v, 16, 32);
}

// ---------------------------------------------------------------------------
// Phase 1: stable sort by depth (bitonic on u64 keys) + precompute sorted
// gaussian feature vectors for the WMMA bilinear sigma evaluation.
// features: [0.5a, 0.5c, b, -(a*mx+b*my), -(c*my+b*mx), 0.5a*mx^2+0.5c*my^2+b*mx*my, 0, 0]
// (coords centered at 64 to keep f32 rounding small)
// ---------------------------------------------------------------------------
__global__ __launch_bounds__(1024)
void pgr_sort_prep(const float* __restrict__ means2d,
                   const float* __restrict__ conics,
                   const float* __restrict__ colors,
                   const float* __restrict__ opacities,
                   const float* __restrict__ depths,
                   float* __restrict__ feat,     // [G][8]
                   float* __restrict__ colop) {  // [G][4] rgb + opacity
    __shared__ unsigned long long skey[GG];
    const int t = threadIdx.x;

    for (int i = t; i < GG; i += 1024) {
        unsigned int bits = __float_as_uint(depths[i]);  // depths > 0: bit order == value order
        skey[i] = ((unsigned long long)bits << 32) | (unsigned int)i;
    }
    __syncthreads();

    for (int k = 2; k <= GG; k <<= 1) {
        for (int j = k >> 1; j > 0; j >>= 1) {
            int low = t & (j - 1);
            int i1  = 2 * t - low;
            int i2  = i1 + j;
            bool up = ((i1 & k) == 0);
            unsigned long long a = skey[i1];
            unsigned long long b = skey[i2];
            if ((a > b) == up) { skey[i1] = b; skey[i2] = a; }
            __syncthreads();
        }
    }

    for (int s = t; s < GG; s += 1024) {
        int g = (int)(skey[s] & 0xFFFFFFFFu);
        float mx = means2d[2 * g + 0] - 64.0f;
        float my = means2d[2 * g + 1] - 64.0f;
        float a  = conics[3 * g + 0];
        float b  = conics[3 * g + 1];
        float c  = conics[3 * g + 2];
        float* f = feat + s * 8;
        f[0] = 0.5f * a;
        f[1] = 0.5f * c;
        f[2] = b;
        f[3] = -(a * mx + b * my);
        f[4] = -(c * my + b * mx);
        f[5] = 0.5f * a * mx * mx + 0.5f * c * my * my + b * mx * my;
        f[6] = 0.0f;
        f[7] = 0.0f;
        float* q = colop + s * 4;
        q[0] = colors[3 * g + 0];
        q[1] = colors[3 * g + 1];
        q[2] = colors[3 * g + 2];
        q[3] = opacities[g];
    }
}

// ---------------------------------------------------------------------------
// Phase 2: raster. One wave handles a 4x4 pixel tile (16 pixels = N dim).
// Per 16-gaussian chunk: 2x V_WMMA_F32_16X16X4_F32 produce sigma[g][p];
// lane p holds gaussians 0..7, lane p+16 holds 8..15 (sequential in VGPRs).
// Raw transmittance is validity-free, so one shuffle of the chunk product
// gives the hi half its true prefix T; weights are thresholded exactly.
// ---------------------------------------------------------------------------
__global__ __launch_bounds__(256)
void pgr_raster(const float* __restrict__ feat,
                const float* __restrict__ colop,
                float* __restrict__ out) {
    const int lane = threadIdx.x & 31;
    const int wave = threadIdx.x >> 5;           // 0..7
    const int p    = lane & 15;                  // pixel slot (N)
    const int half = lane >> 4;                  // 0: gauss 0..7, 1: gauss 8..15
    const int k0   = half * 2;                   // K pair for A/B operands

    const int bx = blockIdx.x & 15;              // 16 blocks across (8 px each)
    const int by = blockIdx.x >> 4;              // 8 blocks down (16 px each)
    const int tile_x0 = bx * 8 + (wave & 1) * 4;
    const int tile_y0 = by * 16 + (wave >> 1) * 4;
    const int px_i = tile_x0 + (p & 3);
    const int py_i = tile_y0 + (p >> 2);
    const float px = (float)px_i + 0.5f - 64.0f;
    const float py = (float)py_i + 0.5f - 64.0f;

    const float pf[8] = { px * px, py * py, px * py, px, py, 1.0f, 0.0f, 0.0f };
    const v2f b_lo = { pf[k0 + 0], pf[k0 + 1] };
    const v2f b_hi = { pf[4 + k0 + 0], pf[4 + k0 + 1] };

    float T = 1.0f;
    float r_acc = 0.0f, g_acc = 0.0f, bl_acc = 0.0f;

    for (int base = 0; base < GG; base += 16) {
        // A operands: lane supplies features k0,k0+1 (and 4+k0,4+k0+1) of gaussian base+p
        const float* fp = feat + (size_t)(base + p) * 8 + k0;
        const v2f a_lo = *(const v2f*)(fp + 0);
        const v2f a_hi = *(const v2f*)(fp + 4);

        v8f c = {};
        c = __builtin_amdgcn_wmma_f32_16x16x4_f32(false, a_lo, false, b_lo,
                                                  (short)0, c, false, false);
        c = __builtin_amdgcn_wmma_f32_16x16x4_f32(false, a_hi, false, b_hi,
                                                  (short)0, c, false, false);

        const int gb = base + 8 * half;
        float alpha[8];
        v4f co[8];
#pragma unroll
        for (int i = 0; i < 8; ++i) {
            co[i] = *(const v4f*)(colop + (size_t)(gb + i) * 4);
            float sig = c[i];
            float a = co[i].w * __expf(-sig);
            a = fminf(a, 0.99f);
            a = (sig < 0.0f || a < (1.0f / 255.0f)) ? 0.0f : a;
            alpha[i] = a;
        }

        // raw chunk transmittance product (validity-free)
        float P = 1.0f;
#pragma unroll
        for (int i = 0; i < 8; ++i) P *= (1.0f - alpha[i]);

        float Pother = shflx16(P);                    // hi receives lo's product
        float Tin = half ? (T * Pother) : T;          // true prefix T for this half

#pragma unroll
        for (int i = 0; i < 8; ++i) {
            float Tn = Tin * (1.0f - alpha[i]);
            float w  = (Tn >= 1e-4f) ? alpha[i] * Tin : 0.0f;
            r_acc  += w * co[i].x;
            g_acc  += w * co[i].y;
            bl_acc += w * co[i].z;
            Tin = Tn;
        }

        float Tend = shflx16(Tin);                    // lo receives hi's final T
        T = half ? Tin : Tend;                        // both halves: chunk-end raw T

        if (__all(T < 1e-4f)) break;                  // wave-uniform -> EXEC stays full
    }

    // combine the two half-lane partial colors for each pixel
    r_acc  += shflx16(r_acc);
    g_acc  += shflx16(g_acc);
    bl_acc += shflx16(bl_acc);
    if (half == 0) {
        float* o = out + ((size_t)py_i * WW + px_i) * 3;
        o[0] = r_acc;
        o[1] = g_acc;
        o[2] = bl_acc;
    }
}

extern "C" void kernel_launch(void* const* d_in, const int* in_sizes, int n_in,
                              void* d_out, int out_size, void* d_ws, size_t ws_size,
                              hipStream_t stream) {
    const float* means2d   = (const float*)d_in[0];
    const float* conics    = (const float*)d_in[1];
    const float* colors    = (const float*)d_in[2];
    const float* opacities = (const float*)d_in[3];
    const float* depths    = (const float*)d_in[4];

    float* feat  = (float*)d_ws;            // 2048*8 floats = 64 KB
    float* colop = feat + (size_t)GG * 8;   // 2048*4 floats = 32 KB

    pgr_sort_prep<<<1, 1024, 0, stream>>>(means2d, conics, colors, opacities,
                                          depths, feat, colop);
    pgr_raster<<<128, 256, 0, stream>>>(feat, colop, (float*)d_out);
}